// ErbNorm_29557964931353
// MI455X (gfx1250) — compile-verified
//
#include <hip/hip_runtime.h>
#include <math.h>

// Problem constants (fixed shapes from the reference)
#define ALPHA_F      0.99f
#define OMA_F        0.01f        // 1 - alpha
#define EPS_F        1e-12f
#define INIT_HI_F   -60.0f
#define INIT_LO_F   -90.0f
#define VAR0_F       1600.0f      // 40^2
#define T_DIM        4000
#define F_DIM        64
#define CHUNK        16
#define NCHUNK       (T_DIM / CHUNK)   // 250
#define NBUF         12                // LDS ring slots per wave
#define NPRE         10                // chunks prefetched ahead (ASYNCcnt depth = 2*NPRE+2)
#define ROWF         20                // floats per LDS row (80 B: bank-conflict-free padding)
#define BUFF         (16 * ROWF)       // floats per ring slot (320)
#define BUFB         (BUFF * 4)        // bytes per ring slot (1280)
#define WPB          4                 // waves per block

typedef float v2f __attribute__((ext_vector_type(2)));
typedef float v8f __attribute__((ext_vector_type(8)));

__device__ __forceinline__ float swapx16(float v) {
  // ds_swizzle SWAPX16: xor=0x10, and=0x1f  -> lane l <-> lane l^16
  return __int_as_float(__builtin_amdgcn_ds_swizzle(__float_as_int(v), 0x401f));
}

// One wave32 owns 16 chains (16 consecutive f of one b) for all T.
// Data streamed global -> LDS with async B128 DMA (ASYNCcnt ring, NPRE chunks deep).
// Per 16-step chunk: Y = L*X via 4x v_wmma_f32_16x16x4_f32 (carry-free),
// mu = fma(alpha^{i+1}, carry, Y); var = serial in-lane scan + linear fix-up.
__global__ __launch_bounds__(128) void erbnorm_kernel(const float* __restrict__ x,
                                                      float* __restrict__ out) {
  extern __shared__ float lds_f[];
  const int lane      = threadIdx.x & 31;
  const int waveInBlk = threadIdx.x >> 5;
  const int wave      = blockIdx.x * WPB + waveInBlk;   // 1024 waves total
  const int chain0    = wave << 4;
  const int b         = chain0 >> 6;                    // F = 64
  const int f0        = chain0 & 63;
  const int c         = lane & 15;                      // chain / N index
  const int g         = lane >> 4;                      // lane half

  const float* sa = x + (size_t)b * T_DIM * F_DIM;      // SGPR base for async loads
  float*       ob = out + (size_t)b * T_DIM * F_DIM + f0 + c;

  // ---- async-load per-lane address terms: lane -> (row = lane/4, 16B slice lane%4)
  const int rowq = lane >> 2;
  const int colq = (lane & 3) * 16;
  const unsigned laneG = (unsigned)(f0 * 4 + rowq * 256 + colq);     // global bytes
  const unsigned laneL = (unsigned)(rowq * 80 + colq);               // LDS bytes
  const int waveBaseF  = waveInBlk * NBUF * BUFF;
  const unsigned ldsWaveByte = (unsigned)(size_t)(lds_f + waveBaseF);

  // ---- A matrix (decay L), per-lane constants. A layout: lane%16=M, K = r+2g in slice s
  float Areg[4][2];
#pragma unroll
  for (int s = 0; s < 4; ++s)
#pragma unroll
    for (int r = 0; r < 2; ++r) {
      const int m = c, k = 4 * s + 2 * g + r;
      Areg[s][r] = (m >= k) ? OMA_F * __powf(ALPHA_F, (float)(m - k)) : 0.0f;
    }

  float scale[8], fixs[8];
  int   offDl[8], offDg[8], offBl[4][2];
#pragma unroll
  for (int r = 0; r < 8; ++r) {
    scale[r] = __powf(ALPHA_F, (float)(r + 8 * g + 1));   // alpha^{M+1}, M = r+8g
    fixs[r]  = g ? __powf(ALPHA_F, (float)(r + 1)) : 0.0f;
    offDl[r] = (r + 8 * g) * ROWF + c;                    // LDS C/D-layout read
    offDg[r] = (r + 8 * g) * F_DIM;                       // global store offset
  }
#pragma unroll
  for (int s = 0; s < 4; ++s)
#pragma unroll
    for (int r = 0; r < 2; ++r)
      offBl[s][r] = (4 * s + 2 * g + r) * ROWF + c;       // LDS B-layout read
  const float A8 = __powf(ALPHA_F, 8.0f);

  // ---- initial carries (replicated across both lane halves)
  const float step = (INIT_LO_F - INIT_HI_F) / (float)(F_DIM - 1);
  float muRepl  = INIT_HI_F + (float)(f0 + c) * step;
  float varRepl = VAR0_F;

  // ---- prologue: fill the async pipeline (2 B128 DMAs = 1 KB per chunk per wave)
#pragma unroll
  for (int k = 0; k < NPRE; ++k) {
    const unsigned v0 = laneG + (unsigned)k * 4096u;
    const unsigned l0 = ldsWaveByte + (unsigned)k * BUFB + laneL;
    asm volatile("global_load_async_to_lds_b128 %0, %1, %2"
                 :: "v"(l0), "v"(v0), "s"(sa) : "memory");
    asm volatile("global_load_async_to_lds_b128 %0, %1, %2"
                 :: "v"(l0 + 640u), "v"(v0 + 2048u), "s"(sa) : "memory");
  }

  int rdSlot = 0, wrSlot = NPRE;
  unsigned gPref = (unsigned)NPRE * 4096u;   // byte offset of chunk being prefetched
  int obase = 0;                             // output float offset of current chunk

  for (int i = 0; i < NCHUNK; ++i) {
    // ---- keep pipeline depth constant (tail re-reads last chunk into unused slots)
    {
      const unsigned v0 = laneG + gPref;
      const unsigned l0 = ldsWaveByte + (unsigned)wrSlot * BUFB + laneL;
      asm volatile("global_load_async_to_lds_b128 %0, %1, %2"
                   :: "v"(l0), "v"(v0), "s"(sa) : "memory");
      asm volatile("global_load_async_to_lds_b128 %0, %1, %2"
                   :: "v"(l0 + 640u), "v"(v0 + 2048u), "s"(sa) : "memory");
    }
    if (i + NPRE + 1 < NCHUNK) gPref += 4096u;
    if (++wrSlot == NBUF) wrSlot = 0;

    // ---- wait: oldest chunk (this one) is resident in LDS
    asm volatile("s_wait_asynccnt %0" :: "i"(2 * NPRE) : "memory");

    const int rb = waveBaseF + rdSlot * BUFF;
    if (++rdSlot == NBUF) rdSlot = 0;

    // ---- LDS reads in both WMMA layouts
    float xB[4][2], xD[8];
#pragma unroll
    for (int s = 0; s < 4; ++s)
#pragma unroll
      for (int r = 0; r < 2; ++r)
        xB[s][r] = lds_f[rb + offBl[s][r]];
#pragma unroll
    for (int r = 0; r < 8; ++r)
      xD[r] = lds_f[rb + offDl[r]];

    // ---- Y = L * X via 4 chained WMMAs (carry-free: off the serial path)
    v8f acc = {};
#pragma unroll
    for (int s = 0; s < 4; ++s) {
      v2f a;  a[0]  = Areg[s][0]; a[1]  = Areg[s][1];
      v2f bb; bb[0] = xB[s][0];   bb[1] = xB[s][1];
      acc = __builtin_amdgcn_wmma_f32_16x16x4_f32(
          false, a, false, bb, (short)0, acc, false, false);
    }

    // ---- mu = Y + alpha^{M+1} * carry ; e = x - mu ; d = e^2
    float e[8], dsq[8], mu7 = 0.0f;
#pragma unroll
    for (int r = 0; r < 8; ++r) {
      const float mu = fmaf(scale[r], muRepl, acc[r]);
      if (r == 7) mu7 = mu;
      e[r]   = xD[r] - mu;
      dsq[r] = e[r] * e[r];
    }

    // ---- var: serial in-lane scan (low half true carry, high half zero carry)
    float v = g ? 0.0f : varRepl;
    float var_[8];
#pragma unroll
    for (int r = 0; r < 8; ++r) {
      v = fmaf(ALPHA_F, v, OMA_F * dsq[r]);
      var_[r] = v;
    }
    // one swizzle serves fix-up AND carry:
    //   low half  P = var7_true, high half P = partial15
    const float P    = v;
    const float swzP = swapx16(P);
    const float var7t  = g ? swzP : P;     // true var at t=7 (replicated)
    const float part15 = g ? P : swzP;     // zero-carry partial at t=15 (replicated)
    varRepl = fmaf(A8, var7t, part15);     // exact var15 -> next-chunk carry
#pragma unroll
    for (int r = 0; r < 8; ++r)
      var_[r] = fmaf(fixs[r], swzP, var_[r]);   // fixs==0 on low half

    // ---- out = e / (sqrt(var)+eps) with fast v_sqrt + v_rcp
#pragma unroll
    for (int r = 0; r < 8; ++r)
      ob[obase + offDg[r]] =
          e[r] * __builtin_amdgcn_rcpf(__builtin_amdgcn_sqrtf(var_[r]) + EPS_F);
    obase += CHUNK * F_DIM;

    // ---- mu carry (mu15 lives in high half, vgpr 7)
    const float swzMu = swapx16(mu7);
    muRepl = g ? mu7 : swzMu;
  }
}

extern "C" void kernel_launch(void* const* d_in, const int* in_sizes, int n_in,
                              void* d_out, int out_size, void* d_ws, size_t ws_size,
                              hipStream_t stream) {
  (void)in_sizes; (void)n_in; (void)d_ws; (void)ws_size; (void)out_size;
  const float* x = (const float*)d_in[0];
  float* out = (float*)d_out;
  // 1024 waves = 16384 chains / 16; 4 waves (128 thr) per block -> 256 blocks
  const size_t shmem = (size_t)WPB * NBUF * BUFB;   // 61440 B
  dim3 grid(256), block(128);
  hipLaunchKernelGGL(erbnorm_kernel, grid, block, shmem, stream, x, out);
}